// PointTransformer_6150393168285
// MI455X (gfx1250) — compile-verified
//
#include <hip/hip_runtime.h>
#include <hip/hip_bf16.h>

// ---------------------------------------------------------------------------
// PointTransformer-style grouped attention for MI455X (gfx1250, wave32).
//
// Roofline: encoder GEMMs ~1.07 GF + refine ~0.27 GF (noise vs WMMA ceiling);
// dominant cost is ~270MB of random 256B-row gathers, fully resident in the
// 192MB L2 => the pipeline is L2-bandwidth bound. Keep f32 precision
// (GEMMs are free) and use V_WMMA_F32_16X16X4_F32 for exact reference
// numerics.
//
// Weights are staged into LDS with the gfx1250 async copy engine using a
// *pair-interleaved* layout: sW2[k/2][n] = {W[k][n], W[k+1][n]}, so every
// WMMA B-operand is one naturally-aligned ds_load_b64 directly into the
// even-aligned VGPR pair the matrix pipe consumes — no repacking v_movs.
// ---------------------------------------------------------------------------

typedef __attribute__((ext_vector_type(2))) float v2f;
typedef __attribute__((ext_vector_type(8))) float v8f;

#define BN_EPS 1e-5f

constexpr int Bc = 4;
constexpr int Nc = 16384;
constexpr int Mc = 4096;
constexpr int Kc = 32;
constexpr int Ci = 64;
constexpr int C0 = 64;
constexpr int Co = 128;

// Async per-lane 4B global->LDS copy. VDST = LDS byte address (low 32 bits of
// the generic pointer = wave-relative LDS address per ISA flat-aperture
// truncation), VADDR = per-lane 64-bit global address. Tracked by ASYNCcnt.
__device__ __forceinline__ void async_g2lds_b32(const float* gptr, float* lptr) {
    const unsigned lds_off = (unsigned)(uintptr_t)lptr;
    asm volatile("global_load_async_to_lds_b32 %0, %1, off"
                 :: "v"(lds_off), "v"(gptr)
                 : "memory");
}

__device__ __forceinline__ void wait_async_all() {
    asm volatile("s_wait_asynccnt 0x0" ::: "memory");
}

// ---------------------------------------------------------------------------
// Kernel 1: feat_feats = BN(feats @ W_feat), attention = BN(feats @ W_attn)
// One wave computes a 16-row tile over all 64 output columns for BOTH
// matrices (A operand shared). f32 WMMA 16x16x4, K-loop of 16 steps.
// W_feat/W_attn async-staged into pair-interleaved LDS (16KB each).
// ---------------------------------------------------------------------------
__global__ __launch_bounds__(256)
void pt_encode_kernel(const float* __restrict__ feats,
                      const float* __restrict__ Wf, const float* __restrict__ Wa,
                      const float* __restrict__ fg, const float* __restrict__ fb,
                      const float* __restrict__ fm, const float* __restrict__ fv,
                      const float* __restrict__ ag, const float* __restrict__ ab,
                      const float* __restrict__ am, const float* __restrict__ av,
                      float* __restrict__ outF, float* __restrict__ outA)
{
    __shared__ v2f sWf2[(Ci / 2) * C0];   // 16KB: [j][n] = {W[2j][n], W[2j+1][n]}
    __shared__ v2f sWa2[(Ci / 2) * C0];   // 16KB

    // Transposing async stage: each lane drops one element into its pair slot.
    // Global side: consecutive lanes -> consecutive n => 256B coalesced.
    #pragma unroll
    for (int i = 0; i < 16; ++i) {
        const int e = i * 256 + threadIdx.x;       // 0..4095
        const int k = e >> 6;                      // row (K index)
        const int n = e & 63;                      // column
        const int slot = ((k >> 1) * C0 + n) * 2 + (k & 1);
        async_g2lds_b32(Wf + e, ((float*)sWf2) + slot);
        async_g2lds_b32(Wa + e, ((float*)sWa2) + slot);
    }
    wait_async_all();      // each wave waits for its own async copies
    __syncthreads();       // then all waves' copies are visible

    const int lane = threadIdx.x & 31;
    const int wave = threadIdx.x >> 5;
    const int r0   = (blockIdx.x * 8 + wave) * 16;   // 16 rows of (B*N, Ci)
    const int lr   = lane & 15;
    const int hi   = lane >> 4;                      // 0 or 1

    v8f accF[4] = {};   // column tiles n0 = 0,16,32,48
    v8f accA[4] = {};

    const float* arow = feats + (size_t)(r0 + lr) * Ci;

    #pragma unroll
    for (int k0 = 0; k0 < Ci; k0 += 4) {
        const v2f a = *(const v2f*)(arow + k0 + 2 * hi);   // A: K=k0..k0+3
        const int j = (k0 >> 1) + hi;                      // pair row
        #pragma unroll
        for (int nt = 0; nt < 4; ++nt) {
            const int n = nt * 16 + lr;
            const v2f bF = sWf2[j * C0 + n];   // single ds_load_b64
            const v2f bA = sWa2[j * C0 + n];
            accF[nt] = __builtin_amdgcn_wmma_f32_16x16x4_f32(
                false, a, false, bF, (short)0, accF[nt], false, false);
            accA[nt] = __builtin_amdgcn_wmma_f32_16x16x4_f32(
                false, a, false, bA, (short)0, accA[nt], false, false);
        }
    }

    // Fold BatchNorm into per-channel scale/shift, store.
    #pragma unroll
    for (int nt = 0; nt < 4; ++nt) {
        const int n = nt * 16 + lr;
        const float sF = fg[n] * rsqrtf(fv[n] + BN_EPS);
        const float tF = fb[n] - fm[n] * sF;
        const float sA = ag[n] * rsqrtf(av[n] + BN_EPS);
        const float tA = ab[n] - am[n] * sA;
        #pragma unroll
        for (int v = 0; v < 8; ++v) {
            const int row = r0 + v + 8 * hi;     // C layout: vgpr v -> M=v (+8 hi half)
            outF[(size_t)row * C0 + n] = accF[nt][v] * sF + tF;
            outA[(size_t)row * C0 + n] = accA[nt][v] * sA + tA;
        }
    }
}

// ---------------------------------------------------------------------------
// Kernel 2: gather + xyz-MLP + online softmax over K + weighted aggregation.
// One 64-thread block per (b,m); thread == channel c. Per-k gathers are
// 64 lanes x 4B contiguous = 256B coalesced rows, L2-resident.
// ---------------------------------------------------------------------------
__global__ __launch_bounds__(64)
void pt_aggregate_kernel(const float* __restrict__ xyz,
                         const float* __restrict__ new_xyz,
                         const float* __restrict__ featF,
                         const float* __restrict__ featA,
                         const float* __restrict__ Wx,
                         const float* __restrict__ xg, const float* __restrict__ xb,
                         const float* __restrict__ xm, const float* __restrict__ xv,
                         const int* __restrict__ gind,
                         float* __restrict__ outAgg)
{
    const int bm = blockIdx.x;              // b*M + m
    const int b  = bm / Mc;
    const int c  = threadIdx.x;             // 0..63 channel

    __shared__ int   s_idx[Kc];
    __shared__ float s_dx[Kc], s_dy[Kc], s_dz[Kc];

    if (threadIdx.x < Kc) {
        const int idx = gind[(size_t)bm * Kc + threadIdx.x];
        s_idx[threadIdx.x] = idx;
        const float* p = xyz + ((size_t)b * Nc + idx) * 3;
        const float* q = new_xyz + (size_t)bm * 3;
        s_dx[threadIdx.x] = p[0] - q[0];
        s_dy[threadIdx.x] = p[1] - q[1];
        s_dz[threadIdx.x] = p[2] - q[2];
    }
    __syncthreads();

    const float w0 = Wx[c];
    const float w1 = Wx[C0 + c];
    const float w2 = Wx[2 * C0 + c];
    const float xs = xg[c] * rsqrtf(xv[c] + BN_EPS);
    const float xt = xb[c] - xm[c] * xs;

    float mv = -INFINITY, se = 0.0f, ws = 0.0f;

    #pragma unroll 4
    for (int k = 0; k < Kc; ++k) {
        const int idx = s_idx[k];
        // prefetch a future gather row into the cache hierarchy
        if (k + 8 < Kc) {
            const int pidx = s_idx[k + 8];
            __builtin_prefetch(featF + ((size_t)b * Nc + pidx) * C0 + c, 0, 0);
            __builtin_prefetch(featA + ((size_t)b * Nc + pidx) * C0 + c, 0, 0);
        }
        const float f  = featF[((size_t)b * Nc + idx) * C0 + c];
        const float at = featA[((size_t)b * Nc + idx) * C0 + c];
        const float xf = (s_dx[k] * w0 + s_dy[k] * w1 + s_dz[k] * w2) * xs + xt;
        const float gf = f + xf;
        const float ga = at + xf;
        // online softmax with fused weighted accumulation
        const float mn   = fmaxf(mv, gf);
        const float corr = __expf(mv - mn);
        const float e    = __expf(gf - mn);
        se = se * corr + e;
        ws = ws * corr + e * ga;
        mv = mn;
    }
    outAgg[(size_t)bm * C0 + c] = ws / se;
}

// ---------------------------------------------------------------------------
// Kernel 3: out = relu(BN(agg @ W_ref)) * (cnt > 0). f32 WMMA, 8 col tiles.
// W_ref async-staged into pair-interleaved LDS (32KB).
// ---------------------------------------------------------------------------
__global__ __launch_bounds__(256)
void pt_refine_kernel(const float* __restrict__ agg,   // (B*M, C0)
                      const float* __restrict__ Wr,    // (C0, Co)
                      const float* __restrict__ rg, const float* __restrict__ rb,
                      const float* __restrict__ rm, const float* __restrict__ rv,
                      const int* __restrict__ cnt,     // (B*M)
                      float* __restrict__ out)         // (B*M, Co)
{
    __shared__ v2f sWr2[(C0 / 2) * Co];   // 32KB: [j][n] = {W[2j][n], W[2j+1][n]}

    #pragma unroll
    for (int i = 0; i < 32; ++i) {
        const int e = i * 256 + threadIdx.x;       // 0..8191
        const int k = e >> 7;                      // row (K index)
        const int n = e & 127;                     // column
        const int slot = ((k >> 1) * Co + n) * 2 + (k & 1);
        async_g2lds_b32(Wr + e, ((float*)sWr2) + slot);
    }
    wait_async_all();
    __syncthreads();

    const int lane = threadIdx.x & 31;
    const int wave = threadIdx.x >> 5;
    const int r0   = (blockIdx.x * 8 + wave) * 16;
    const int lr   = lane & 15;
    const int hi   = lane >> 4;

    v8f acc[8] = {};
    const float* arow = agg + (size_t)(r0 + lr) * C0;

    #pragma unroll
    for (int k0 = 0; k0 < C0; k0 += 4) {
        const v2f a = *(const v2f*)(arow + k0 + 2 * hi);
        const int j = (k0 >> 1) + hi;
        #pragma unroll
        for (int nt = 0; nt < 8; ++nt) {
            const int n = nt * 16 + lr;
            const v2f bv = sWr2[j * Co + n];   // single ds_load_b64
            acc[nt] = __builtin_amdgcn_wmma_f32_16x16x4_f32(
                false, a, false, bv, (short)0, acc[nt], false, false);
        }
    }

    #pragma unroll
    for (int nt = 0; nt < 8; ++nt) {
        const int n = nt * 16 + lr;
        const float s = rg[n] * rsqrtf(rv[n] + BN_EPS);
        const float t = rb[n] - rm[n] * s;
        #pragma unroll
        for (int v = 0; v < 8; ++v) {
            const int row = r0 + v + 8 * hi;
            const float val = fmaxf(acc[nt][v] * s + t, 0.0f);
            const float msk = (cnt[row] > 0) ? 1.0f : 0.0f;
            out[(size_t)row * Co + n] = val * msk;
        }
    }
}

// ---------------------------------------------------------------------------
// Host-side launcher (graph-capture safe: only kernel launches on `stream`).
// ---------------------------------------------------------------------------
extern "C" void kernel_launch(void* const* d_in, const int* in_sizes, int n_in,
                              void* d_out, int out_size, void* d_ws, size_t ws_size,
                              hipStream_t stream) {
    (void)in_sizes; (void)n_in; (void)out_size; (void)ws_size;

    const float* xyz      = (const float*)d_in[0];
    const float* new_xyz  = (const float*)d_in[1];
    const float* feats    = (const float*)d_in[2];
    const float* W_feat   = (const float*)d_in[3];
    const float* fg       = (const float*)d_in[4];
    const float* fb       = (const float*)d_in[5];
    const float* fm       = (const float*)d_in[6];
    const float* fv       = (const float*)d_in[7];
    const float* W_attn   = (const float*)d_in[8];
    const float* ag       = (const float*)d_in[9];
    const float* ab       = (const float*)d_in[10];
    const float* am       = (const float*)d_in[11];
    const float* av       = (const float*)d_in[12];
    const float* W_xyz    = (const float*)d_in[13];
    const float* xg       = (const float*)d_in[14];
    const float* xb       = (const float*)d_in[15];
    const float* xm       = (const float*)d_in[16];
    const float* xv       = (const float*)d_in[17];
    const float* W_ref    = (const float*)d_in[18];
    const float* rg       = (const float*)d_in[19];
    const float* rb       = (const float*)d_in[20];
    const float* rm       = (const float*)d_in[21];
    const float* rv       = (const float*)d_in[22];
    const int*   gind     = (const int*)d_in[23];
    const int*   gcnt     = (const int*)d_in[24];

    float* out = (float*)d_out;

    // Workspace layout (floats): featF | featA | agg  => ~36 MB total
    float* wsF   = (float*)d_ws;
    float* featF = wsF;                                   // B*N*C0
    float* featA = wsF + (size_t)Bc * Nc * C0;            // B*N*C0
    float* aggW  = wsF + 2 * (size_t)Bc * Nc * C0;        // B*M*C0

    // 1) encoder GEMMs + BN: (B*N,Ci)@(Ci,C0) x2, 16 rows/wave, 8 waves/block
    {
        const int blocks = (Bc * Nc) / (16 * 8);          // 512
        pt_encode_kernel<<<blocks, 256, 0, stream>>>(
            feats, W_feat, W_attn,
            fg, fb, fm, fv, ag, ab, am, av,
            featF, featA);
    }

    // 2) gather + xyz MLP + softmax aggregation: one block per (b,m)
    {
        const int blocks = Bc * Mc;                        // 16384
        pt_aggregate_kernel<<<blocks, 64, 0, stream>>>(
            xyz, new_xyz, featF, featA,
            W_xyz, xg, xb, xm, xv,
            gind, aggW);
    }

    // 3) refine GEMM + BN + ReLU + empty mask: (B*M,C0)@(C0,Co)
    {
        const int blocks = (Bc * Mc) / (16 * 8);           // 128
        pt_refine_kernel<<<blocks, 256, 0, stream>>>(
            aggW, W_ref, rg, rb, rm, rv, gcnt, out);
    }
}